// ForwardDiffusedSkinning_31585189494876
// MI455X (gfx1250) — compile-verified
//
#include <hip/hip_runtime.h>

#define NJ 24

typedef __attribute__((ext_vector_type(2))) float v2f;
typedef __attribute__((ext_vector_type(8))) float v8f;

// ---------------- Kernel 1: joint transforms (tiny) ----------------

__device__ inline void rodrigues3(const float* rv, float* Rm) {
    float x = rv[0], y = rv[1], z = rv[2];
    float ang = sqrtf(x * x + y * y + z * z) + 1e-8f;
    float kx = x / ang, ky = y / ang, kz = z / ang;
    float s = sinf(ang), c = cosf(ang), t = 1.0f - c;
    Rm[0] = 1.0f + t * (-(ky * ky + kz * kz));
    Rm[1] = s * (-kz) + t * (kx * ky);
    Rm[2] = s * (ky) + t * (kx * kz);
    Rm[3] = s * (kz) + t * (kx * ky);
    Rm[4] = 1.0f + t * (-(kx * kx + kz * kz));
    Rm[5] = s * (-kx) + t * (ky * kz);
    Rm[6] = s * (-ky) + t * (kx * kz);
    Rm[7] = s * (kx) + t * (ky * kz);
    Rm[8] = 1.0f + t * (-(kx * kx + ky * ky));
}

__device__ inline void mat4mul(const float* A, const float* B, float* C) {
    for (int r = 0; r < 4; ++r)
        for (int c = 0; c < 4; ++c) {
            float acc = 0.0f;
            for (int k = 0; k < 4; ++k) acc += A[r * 4 + k] * B[k * 4 + c];
            C[r * 4 + c] = acc;
        }
}

__device__ inline void build_chain(const float* poses_j3, const float* rest,
                                   const int* parents, float chain[NJ][16]) {
    for (int j = 0; j < NJ; ++j) {
        float Rm[9];
        rodrigues3(poses_j3 + j * 3, Rm);
        float tx = rest[j * 3 + 0], ty = rest[j * 3 + 1], tz = rest[j * 3 + 2];
        int p = parents[j];
        if (p >= 0) { tx -= rest[p * 3 + 0]; ty -= rest[p * 3 + 1]; tz -= rest[p * 3 + 2]; }
        float T[16] = {Rm[0], Rm[1], Rm[2], tx,
                       Rm[3], Rm[4], Rm[5], ty,
                       Rm[6], Rm[7], Rm[8], tz,
                       0.f, 0.f, 0.f, 1.f};
        if (p < 0) {
            for (int e = 0; e < 16; ++e) chain[j][e] = T[e];
        } else {
            mat4mul(chain[p], T, chain[j]);
        }
    }
}

__device__ inline void make_rel(const float* ch, const float* rj, float* rel) {
    float rx = rj[0], ry = rj[1], rz = rj[2];
    for (int r = 0; r < 3; ++r) {
        float tj = ch[r * 4 + 0] * rx + ch[r * 4 + 1] * ry + ch[r * 4 + 2] * rz;
        rel[r * 4 + 0] = ch[r * 4 + 0];
        rel[r * 4 + 1] = ch[r * 4 + 1];
        rel[r * 4 + 2] = ch[r * 4 + 2];
        rel[r * 4 + 3] = ch[r * 4 + 3] - tj;
    }
    rel[12] = 0.f; rel[13] = 0.f; rel[14] = 0.f; rel[15] = 1.f;
}

__device__ inline void affine_inv(const float* M, float* Minv) {
    float a = M[0], b = M[1], c = M[2];
    float d = M[4], e = M[5], f = M[6];
    float g = M[8], h = M[9], i = M[10];
    float A_ = e * i - f * h, B_ = -(d * i - f * g), C_ = d * h - e * g;
    float det = a * A_ + b * B_ + c * C_;
    float id = 1.0f / det;
    float i00 = A_ * id, i01 = -(b * i - c * h) * id, i02 = (b * f - c * e) * id;
    float i10 = B_ * id, i11 = (a * i - c * g) * id, i12 = -(a * f - c * d) * id;
    float i20 = C_ * id, i21 = -(a * h - b * g) * id, i22 = (a * e - b * d) * id;
    float tx = M[3], ty = M[7], tz = M[11];
    Minv[0] = i00; Minv[1] = i01; Minv[2] = i02; Minv[3] = -(i00 * tx + i01 * ty + i02 * tz);
    Minv[4] = i10; Minv[5] = i11; Minv[6] = i12; Minv[7] = -(i10 * tx + i11 * ty + i12 * tz);
    Minv[8] = i20; Minv[9] = i21; Minv[10] = i22; Minv[11] = -(i20 * tx + i21 * ty + i22 * tz);
    Minv[12] = 0.f; Minv[13] = 0.f; Minv[14] = 0.f; Minv[15] = 1.f;
}

__global__ void tfs_kernel(const float* __restrict__ poses,
                           const float* __restrict__ cano_pose,
                           const float* __restrict__ rest,
                           float* __restrict__ rel_out, int B) {
    int bidx = threadIdx.x;
    if (bidx >= B) return;
    const int parents[NJ] = {-1, 0, 0, 0, 1, 2, 3, 4, 5, 6, 7, 8,
                             9, 9, 9, 12, 13, 14, 16, 17, 18, 19, 20, 21};
    float chain[NJ][16];
    float caninv[NJ][16];
    // canonical transforms -> inverse per joint (redundant per thread, tiny)
    build_chain(cano_pose, rest, parents, chain);
    for (int j = 0; j < NJ; ++j) {
        float relc[16];
        make_rel(chain[j], rest + j * 3, relc);
        affine_inv(relc, caninv[j]);
    }
    // batch transforms
    build_chain(poses + bidx * NJ * 3, rest, parents, chain);
    for (int j = 0; j < NJ; ++j) {
        float relb[16], relf[16];
        make_rel(chain[j], rest + j * 3, relb);
        mat4mul(relb, caninv[j], relf);
        for (int e = 0; e < 16; ++e) rel_out[(bidx * NJ + j) * 16 + e] = relf[e];
    }
}

// ---------------- Kernel 2: weight query + WMMA skinning ----------------

__device__ inline void sample24(const float* __restrict__ g, int R, size_t R3,
                                float cx, float cy, float cz, float* w) {
    float Rm1 = (float)(R - 1);
    float Rf = (float)R;
    float gx = fminf(fmaxf(((cx + 1.0f) * Rf - 1.0f) * 0.5f, 0.0f), Rm1);
    float gy = fminf(fmaxf(((cy + 1.0f) * Rf - 1.0f) * 0.5f, 0.0f), Rm1);
    float gz = fminf(fmaxf(((cz + 1.0f) * Rf - 1.0f) * 0.5f, 0.0f), Rm1);
    float x0f = floorf(gx), y0f = floorf(gy), z0f = floorf(gz);
    float fx = gx - x0f, fy = gy - y0f, fz = gz - z0f;
    int x0 = (int)x0f, y0 = (int)y0f, z0 = (int)z0f;
    int x1 = min(x0 + 1, R - 1), y1 = min(y0 + 1, R - 1), z1 = min(z0 + 1, R - 1);
    float wx0 = 1.0f - fx, wx1 = fx;
    float wy0 = 1.0f - fy, wy1 = fy;
    float wz0 = 1.0f - fz, wz1 = fz;

    size_t o000 = ((size_t)z0 * R + y0) * R + x0;
    size_t o001 = ((size_t)z0 * R + y0) * R + x1;
    size_t o010 = ((size_t)z0 * R + y1) * R + x0;
    size_t o011 = ((size_t)z0 * R + y1) * R + x1;
    size_t o100 = ((size_t)z1 * R + y0) * R + x0;
    size_t o101 = ((size_t)z1 * R + y0) * R + x1;
    size_t o110 = ((size_t)z1 * R + y1) * R + x0;
    size_t o111 = ((size_t)z1 * R + y1) * R + x1;
    float w000 = wz0 * wy0 * wx0, w001 = wz0 * wy0 * wx1;
    float w010 = wz0 * wy1 * wx0, w011 = wz0 * wy1 * wx1;
    float w100 = wz1 * wy0 * wx0, w101 = wz1 * wy0 * wx1;
    float w110 = wz1 * wy1 * wx0, w111 = wz1 * wy1 * wx1;

#pragma unroll 4
    for (int c = 0; c < NJ; ++c) {
        const float* gc = g + (size_t)c * R3;
        w[c] = gc[o000] * w000 + gc[o001] * w001 + gc[o010] * w010 + gc[o011] * w011 +
               gc[o100] * w100 + gc[o101] * w101 + gc[o110] * w110 + gc[o111] * w111;
    }
}

__global__ __launch_bounds__(256) void skin_kernel(
    const float* __restrict__ verts,
    const float* __restrict__ wg, const float* __restrict__ lg, const float* __restrict__ rg,
    const float* __restrict__ rel,  // B*NJ*16
    const float* __restrict__ bbe, const float* __restrict__ bbc,
    const float* __restrict__ lhe, const float* __restrict__ lhc,
    const float* __restrict__ rhe, const float* __restrict__ rhc,
    float* __restrict__ out, int N, long total, int R) {
    __shared__ float lds[8 * 768];  // per-wave: 32 verts x 24 weights (reused as 32x16 wt tile)

    const size_t R3 = (size_t)R * R * R;
    int tid = threadIdx.x;
    int wv = tid >> 5;
    int lane = tid & 31;
    float* W = &lds[wv * 768];

    long idx = (long)blockIdx.x * 256 + tid;
    if (idx >= total) idx = total - 1;  // clamp: keeps EXEC all-ones for WMMA
    int b = (int)(idx / N);

    float vx = verts[idx * 3 + 0];
    float vy = verts[idx * 3 + 1];
    float vz = verts[idx * 3 + 2];

    float inv_be = 2.0f / bbe[0];
    float px = (vx - bbc[0]) * inv_be;
    float py = (vy - bbc[1]) * inv_be;
    float pz = (vz - bbc[2]) * inv_be;

    float w24[NJ];
    sample24(wg, R, R3, px, py, pz, w24);

    float le = lhe[0];
    float plx = px * le + lhc[0], ply = py * le + lhc[1], plz = pz * le + lhc[2];
    if (plx >= -1.0f && plx <= 1.0f && ply >= -1.0f && ply <= 1.0f &&
        plz >= -1.0f && plz <= 1.0f) {
        sample24(lg, R, R3, plx, ply, plz, w24);
    }
    float re = rhe[0];
    float prx = px * re + rhc[0], pry = py * re + rhc[1], prz = pz * re + rhc[2];
    if (prx >= -1.0f && prx <= 1.0f && pry >= -1.0f && pry <= 1.0f &&
        prz >= -1.0f && prz <= 1.0f) {
        sample24(rg, R, R3, prx, pry, prz, w24);
    }

    // Stage weights: W[vert_local * 24 + joint]
#pragma unroll
    for (int j = 0; j < NJ; ++j) W[lane * NJ + j] = w24[j];
    __syncthreads();

    // WMMA: (16 verts x 24 joints) @ (24 joints x 16 elems) via 6x K=4 steps.
    // A layout (16x4 f32): lanes hold M=lane&15; vreg0=K{0,2}, vreg1=K{1,3}.
    // B layout (4x16 f32): vreg0=rows K{0,2}, vreg1=rows K{1,3}, N=lane&15.
    const float* relb = rel + (size_t)b * NJ * 16;
    int half = lane >> 4;  // 0: lanes 0-15, 1: lanes 16-31
    int ln = lane & 15;
    v8f acc0 = {};
    v8f acc1 = {};
#pragma unroll
    for (int kb = 0; kb < 6; ++kb) {
        int k0 = 4 * kb + 2 * half;
        v2f bf, a0, a1;
        bf.x = relb[k0 * 16 + ln];
        bf.y = relb[(k0 + 1) * 16 + ln];
        a0.x = W[ln * NJ + k0];
        a0.y = W[ln * NJ + k0 + 1];
        a1.x = W[(16 + ln) * NJ + k0];
        a1.y = W[(16 + ln) * NJ + k0 + 1];
        acc0 = __builtin_amdgcn_wmma_f32_16x16x4_f32(false, a0, false, bf, (short)0, acc0,
                                                     false, false);
        acc1 = __builtin_amdgcn_wmma_f32_16x16x4_f32(false, a1, false, bf, (short)0, acc1,
                                                     false, false);
    }
    __syncthreads();  // done reading weights; reuse W as 32x16 wt tile

    // D layout: vreg r holds rows M=r (lanes 0-15) and M=r+8 (lanes 16-31), N=lane&15
#pragma unroll
    for (int r = 0; r < 8; ++r) {
        int m = r + 8 * half;
        W[m * 16 + ln] = acc0[r];
        W[(16 + m) * 16 + ln] = acc1[r];
    }
    __syncthreads();

    float wt[12];
#pragma unroll
    for (int e = 0; e < 12; ++e) wt[e] = W[lane * 16 + e];
    float ox = wt[0] * vx + wt[1] * vy + wt[2] * vz + wt[3];
    float oy = wt[4] * vx + wt[5] * vy + wt[6] * vz + wt[7];
    float oz = wt[8] * vx + wt[9] * vy + wt[10] * vz + wt[11];
    out[idx * 3 + 0] = ox;
    out[idx * 3 + 1] = oy;
    out[idx * 3 + 2] = oz;
}

// ---------------- Launch ----------------

extern "C" void kernel_launch(void* const* d_in, const int* in_sizes, int n_in,
                              void* d_out, int out_size, void* d_ws, size_t ws_size,
                              hipStream_t stream) {
    const float* verts = (const float*)d_in[0];
    const float* poses = (const float*)d_in[1];
    const float* cano = (const float*)d_in[2];
    const float* restj = (const float*)d_in[3];
    const float* wg = (const float*)d_in[4];
    const float* lg = (const float*)d_in[5];
    const float* rg = (const float*)d_in[6];
    const float* bbe = (const float*)d_in[7];
    const float* bbc = (const float*)d_in[8];
    const float* lhe = (const float*)d_in[9];
    const float* lhc = (const float*)d_in[10];
    const float* rhe = (const float*)d_in[11];
    const float* rhc = (const float*)d_in[12];
    float* out = (float*)d_out;
    float* rel_ws = (float*)d_ws;  // B*24*16 floats

    int B = in_sizes[1] / (NJ * 3);
    int N = in_sizes[0] / (B * 3);
    // grid resolution: in_sizes[4] == NJ * R^3
    long per_ch = (long)in_sizes[4] / NJ;
    int R = 1;
    while ((long)(R + 1) * (R + 1) * (R + 1) <= per_ch) ++R;

    tfs_kernel<<<1, 32, 0, stream>>>(poses, cano, restj, rel_ws, B);

    long total = (long)B * N;
    int blocks = (int)((total + 255) / 256);
    skin_kernel<<<blocks, 256, 0, stream>>>(verts, wg, lg, rg, rel_ws,
                                            bbe, bbc, lhe, lhc, rhe, rhc,
                                            out, N, total, R);
}